// SimpleModel2_55937654063720
// MI455X (gfx1250) — compile-verified
//
#include <hip/hip_runtime.h>
#include <hip/hip_bf16.h>
#include <math.h>

typedef __attribute__((ext_vector_type(2))) float v2f;
typedef __attribute__((ext_vector_type(8))) float v8f;

// ---------------------------------------------------------------------------
// Kernel 1: collapse the scalar-path weight chain into one padded matrix.
//   M = (W1a[8x6] · W1b[6x6] · W2a[6x6] · W2b[6x6]) * 1/(sqrt8 * 6*sqrt6)
// Stored as Mpad[8][16] (cols 6..15 zero) in d_ws for the WMMA B-matrix.
// ---------------------------------------------------------------------------
__global__ void prep_weights_kernel(const float* __restrict__ W1a,
                                    const float* __restrict__ W1b,
                                    const float* __restrict__ W2a,
                                    const float* __restrict__ W2b,
                                    float* __restrict__ Mpad) {
  __shared__ float T1[8 * 6];
  __shared__ float T2[8 * 6];
  const int t = threadIdx.x;           // blockDim.x == 128
  if (t < 48) {                        // T1 = W1a · W1b   [8x6]
    const int c = t / 6, o = t % 6;
    float acc = 0.0f;
    #pragma unroll
    for (int k = 0; k < 6; ++k) acc += W1a[c * 6 + k] * W1b[k * 6 + o];
    T1[c * 6 + o] = acc;
  }
  __syncthreads();
  if (t < 48) {                        // T2 = T1 · W2a    [8x6]
    const int c = t / 6, o = t % 6;
    float acc = 0.0f;
    #pragma unroll
    for (int k = 0; k < 6; ++k) acc += T1[c * 6 + k] * W2a[k * 6 + o];
    T2[c * 6 + o] = acc;
  }
  __syncthreads();
  if (t < 48) {                        // T1 = T2 · W2b * scale   [8x6]
    const int c = t / 6, o = t % 6;
    float acc = 0.0f;
    #pragma unroll
    for (int k = 0; k < 6; ++k) acc += T2[c * 6 + k] * W2b[k * 6 + o];
    const float scale = 1.0f / (sqrtf(8.0f) * 6.0f * sqrtf(6.0f));
    T1[c * 6 + o] = acc * scale;
  }
  __syncthreads();
  // Pad to [8][16] for the WMMA B-matrix.
  const int k = t / 16, n = t % 16;    // t in [0,128)
  Mpad[t] = (n < 6) ? T1[k * 6 + n] : 0.0f;
}

// ---------------------------------------------------------------------------
// Kernel 2: streaming RBF + WMMA tile matmul.
//   Block = 512 threads = 16 waves; wave w owns batch row b = blk*16 + w.
//   Each lane loads 4 points (3x float4, fully coalesced 1536B/row/wave),
//   wave-reduces the centroid, evaluates 8 Gaussians per point, reduces the
//   8 means into LDS. Wave 0 then runs 2x V_WMMA_F32_16X16X4_F32 to apply
//   the fused 8x6 weight matrix to all 16 rows at once.
// ---------------------------------------------------------------------------
__global__ __launch_bounds__(512) void rbf_mix_kernel(
    const float* __restrict__ pos,     // [B,128,3]
    const float* __restrict__ Mpad,    // [8][16]
    float* __restrict__ out) {         // [B,6]
  __shared__ float s_s[16 * 8];        // per-row mean RBF features

  const int lane = threadIdx.x & 31;
  const int wave = threadIdx.x >> 5;
  const long long b = (long long)blockIdx.x * 16 + wave;

  const float* p = pos + b * 384 + lane * 12;   // 4 points per lane
  const float4 q0 = *(const float4*)(p);
  const float4 q1 = *(const float4*)(p + 4);
  const float4 q2 = *(const float4*)(p + 8);
  // points: (q0.x,q0.y,q0.z) (q0.w,q1.x,q1.y) (q1.z,q1.w,q2.x) (q2.y,q2.z,q2.w)

  // Centroid: per-lane partial sums, then wave32 butterfly reduction.
  float sx = q0.x + q0.w + q1.z + q2.y;
  float sy = q0.y + q1.x + q1.w + q2.z;
  float sz = q0.z + q1.y + q2.x + q2.w;
  #pragma unroll
  for (int m = 16; m >= 1; m >>= 1) {
    sx += __shfl_xor(sx, m, 32);
    sy += __shfl_xor(sy, m, 32);
    sz += __shfl_xor(sz, m, 32);
  }
  const float mx = sx * (1.0f / 128.0f);
  const float my = sy * (1.0f / 128.0f);
  const float mz = sz * (1.0f / 128.0f);

  float acc[8] = {0.f, 0.f, 0.f, 0.f, 0.f, 0.f, 0.f, 0.f};
  auto do_point = [&](float x, float y, float z) {
    x -= mx; y -= my; z -= mz;
    const float r = sqrtf(x * x + y * y + z * z + 1e-12f);
    #pragma unroll
    for (int c = 0; c < 8; ++c) {
      const float t = r - (float)c * (3.0f / 7.0f);
      acc[c] += __expf(-t * t);
    }
  };
  do_point(q0.x, q0.y, q0.z);
  do_point(q0.w, q1.x, q1.y);
  do_point(q1.z, q1.w, q2.x);
  do_point(q2.y, q2.z, q2.w);

  #pragma unroll
  for (int c = 0; c < 8; ++c) {
    #pragma unroll
    for (int m = 16; m >= 1; m >>= 1) acc[c] += __shfl_xor(acc[c], m, 32);
  }
  if (lane == 0) {
    #pragma unroll
    for (int c = 0; c < 8; ++c) s_s[wave * 8 + c] = acc[c] * (1.0f / 128.0f);
  }
  __syncthreads();

  if (wave == 0) {
    // A-matrix 16x4 f32 layout (ISA 7.12.2): lane m=L&15; VGPR0 holds K=k0,
    // VGPR1 holds K=k0+1, with k0 = (L<16 ? 0 : 2).  Two WMMAs cover K=0..7.
    const int m  = lane & 15;
    const int k0 = (lane < 16) ? 0 : 2;
    v2f aLo, aHi, bLo, bHi;
    aLo.x = s_s[m * 8 + k0];
    aLo.y = s_s[m * 8 + k0 + 1];
    aHi.x = s_s[m * 8 + 4 + k0];
    aHi.y = s_s[m * 8 + 4 + k0 + 1];
    const int n = lane & 15;           // B: row K striped across lanes per VGPR
    bLo.x = Mpad[(k0) * 16 + n];
    bLo.y = Mpad[(k0 + 1) * 16 + n];
    bHi.x = Mpad[(k0 + 4) * 16 + n];
    bHi.y = Mpad[(k0 + 5) * 16 + n];

    v8f d = {0.f, 0.f, 0.f, 0.f, 0.f, 0.f, 0.f, 0.f};
    d = __builtin_amdgcn_wmma_f32_16x16x4_f32(false, aLo, false, bLo,
                                              (short)0, d, false, false);
    d = __builtin_amdgcn_wmma_f32_16x16x4_f32(false, aHi, false, bHi,
                                              (short)0, d, false, false);

    // C/D layout: VGPR j = rows M=j (lanes 0-15) and M=j+8 (lanes 16-31).
    if (n < 6) {
      const int rowOff = (lane < 16) ? 0 : 8;
      const long long base = (long long)blockIdx.x * 16;
      #pragma unroll
      for (int j = 0; j < 8; ++j)
        out[(base + j + rowOff) * 6 + n] = d[j];
    }
  }
}

// ---------------------------------------------------------------------------
extern "C" void kernel_launch(void* const* d_in, const int* in_sizes, int n_in,
                              void* d_out, int out_size, void* d_ws, size_t ws_size,
                              hipStream_t stream) {
  const float* positions = (const float*)d_in[0];
  const float* W1a = (const float*)d_in[1];
  const float* W1b = (const float*)d_in[2];
  // d_in[3] = W1c, d_in[4] = W1d, d_in[7] = W2c: dead paths in the reference.
  const float* W2a = (const float*)d_in[5];
  const float* W2b = (const float*)d_in[6];
  float* out  = (float*)d_out;
  float* Mpad = (float*)d_ws;          // 128 floats = 512 B scratch

  const long long B = (long long)in_sizes[0] / (128 * 3);   // 32768

  prep_weights_kernel<<<1, 128, 0, stream>>>(W1a, W1b, W2a, W2b, Mpad);
  rbf_mix_kernel<<<(int)(B / 16), 512, 0, stream>>>(positions, Mpad, out);
}